// PhysicsConsistencyModule_47201690583166
// MI455X (gfx1250) — compile-verified
//
#include <hip/hip_runtime.h>
#include <cstdint>

// Spatially-varying defocus blur (gather formulation).
// One pass: per pixel pick its CoC bin, apply that plane's separable Gaussian
// over an LDS-staged tile+halo. Halo staging uses the CDNA5 async
// global->LDS path (global_load_async_to_lds_b32 + s_wait_asynccnt).

#define MAX_COC_F   50.0f
#define NUM_PLANES  32
#define KERNEL_CAP  31
#define RADIUS      15            // KERNEL_CAP / 2 (max halo)
#define TILE        32            // output tile edge
#define HALOW       (TILE + 2*RADIUS)   // 62
#define B_          4
#define C_          3
#define H_          512
#define W_          512
#define NTHREADS    256

static __device__ __forceinline__ int iclamp(int v, int lo, int hi) {
    return v < lo ? lo : (v > hi ? hi : v);
}

__global__ __launch_bounds__(NTHREADS)
void defocus_blur_kernel(const float* __restrict__ img,   // [B,C,H,W]
                         const float* __restrict__ coc,   // [B,1,H,W]
                         float* __restrict__ out)         // [B,C,H,W]
{
    // Planar channel layout: per-channel array, per-lane stride-1 x accesses
    // -> conflict-free across the 64 LDS banks.
    __shared__ float sTile[C_][HALOW * HALOW];            // 46128 B
    __shared__ float sG[NUM_PLANES][32];                  //  4096 B (1D gaussians)
    __shared__ int   sRad[NUM_PLANES];                    //   128 B

    const int tid = threadIdx.x;
    const int x0  = blockIdx.x * TILE;
    const int y0  = blockIdx.y * TILE;
    const int b   = blockIdx.z;

    // ---- (a) rebuild static per-plane 1D Gaussian table (32 threads) ----
    if (tid < NUM_PLANES) {
        const float pcoc = (float)tid * (MAX_COC_F / (float)(NUM_PLANES - 1));
        int r = 0;
        if (pcoc >= 0.5f) {
            int k = (int)(2.0f * pcoc + 1.0f);
            if ((k & 1) == 0) k += 1;
            if (k > KERNEL_CAP) k = KERNEL_CAP;
            r = k >> 1;
            const float sigma  = pcoc / 2.355f;
            const float inv2s2 = 1.0f / (2.0f * sigma * sigma);
            float sum = 0.0f;
            for (int j = 0; j <= 2 * r; ++j) {
                const float d = (float)(j - r);
                const float v = __expf(-d * d * inv2s2);
                sG[tid][j] = v;
                sum += v;
            }
            const float inv = 1.0f / sum;
            for (int j = 0; j <= 2 * r; ++j) sG[tid][j] *= inv;
        } else {
            sG[tid][0] = 1.0f;    // plane 0: identity (kernel None)
        }
        sRad[tid] = r;
    }

    // ---- (b) zero-fill tile: conv zero-padding for out-of-image halo ----
    float* flat = &sTile[0][0];
    for (int i = tid; i < C_ * HALOW * HALOW; i += NTHREADS)
        flat[i] = 0.0f;

    __syncthreads();   // zeros + gaussian table visible block-wide

    // ---- (c) async global -> LDS stage of tile + halo (gfx1250 path) ----
    // Each LDS slot maps to a unique (c, gy, gx). In-bounds lanes issue an
    // async B32 transfer; out-of-bounds slots keep their zero fill.
    {
        const uint64_t sbase = (uint64_t)(uintptr_t)img;   // SGPR pair base
        for (int i = tid; i < C_ * HALOW * HALOW; i += NTHREADS) {
            const int c   = i / (HALOW * HALOW);
            const int rem = i - c * (HALOW * HALOW);
            const int ty  = rem / HALOW;
            const int tx  = rem - ty * HALOW;
            const int gy  = y0 - RADIUS + ty;
            const int gx  = x0 - RADIUS + tx;
            if (gy >= 0 && gy < H_ && gx >= 0 && gx < W_) {
                // flat LDS addresses carry the wave-relative LDS offset in
                // addr[31:0] (ISA 10.2 aperture rules) -> low 32 bits.
                const uint32_t loff = (uint32_t)(uintptr_t)&flat[i];
                const uint32_t goff =
                    (uint32_t)(((((b * C_ + c) * H_) + gy) * W_ + gx) * 4);
                asm volatile(
                    "global_load_async_to_lds_b32 %0, %1, %2"
                    :
                    : "v"(loff), "v"(goff), "s"(sbase)
                    : "memory");
            }
        }
        asm volatile("s_wait_asynccnt 0" ::: "memory");  // my transfers done
    }
    __syncthreads();   // all waves' async transfers done

    // ---- (d) per-pixel gather: bin CoC, apply that plane's 2D gaussian ----
    for (int pix = tid; pix < TILE * TILE; pix += NTHREADS) {
        const int py = pix >> 5;        // TILE == 32
        const int px = pix & 31;
        const int gy = y0 + py;
        const int gx = x0 + px;

        const float cv = coc[(b * H_ + gy) * W_ + gx];
        // Bins are midpoints of uniform linspace -> rounded scale.
        int p = (int)floorf(cv * ((float)(NUM_PLANES - 1) / MAX_COC_F) + 0.5f);
        p = iclamp(p, 0, NUM_PLANES - 1);

        const int r = sRad[p];
        const float* __restrict__ g = sG[p];

        float a0 = 0.0f, a1 = 0.0f, a2 = 0.0f;
        const int base = (py + RADIUS - r) * HALOW + (px + RADIUS - r);
        for (int dy = 0; dy <= 2 * r; ++dy) {
            const float wy = g[dy];
            const int rowOff = base + dy * HALOW;
            for (int dx = 0; dx <= 2 * r; ++dx) {
                const float w = wy * g[dx];
                a0 = fmaf(w, sTile[0][rowOff + dx], a0);
                a1 = fmaf(w, sTile[1][rowOff + dx], a1);
                a2 = fmaf(w, sTile[2][rowOff + dx], a2);
            }
        }

        const size_t o = ((size_t)(b * C_) * H_ + gy) * W_ + gx;
        out[o]                        = a0;
        out[o + (size_t)H_ * W_]      = a1;
        out[o + (size_t)2 * H_ * W_]  = a2;
    }
}

extern "C" void kernel_launch(void* const* d_in, const int* in_sizes, int n_in,
                              void* d_out, int out_size, void* d_ws, size_t ws_size,
                              hipStream_t stream) {
    (void)in_sizes; (void)n_in; (void)out_size; (void)d_ws; (void)ws_size;
    const float* img = (const float*)d_in[0];   // sharp_image [4,3,512,512] f32
    const float* coc = (const float*)d_in[1];   // coc_map     [4,1,512,512] f32
    float* out = (float*)d_out;                 // [4,3,512,512] f32

    dim3 grid(W_ / TILE, H_ / TILE, B_);        // 16 x 16 x 4 workgroups
    defocus_blur_kernel<<<grid, NTHREADS, 0, stream>>>(img, coc, out);
}